// ComputePrivacyBucketsDelta_29274497089611
// MI455X (gfx1250) — compile-verified
//
#include <hip/hip_runtime.h>

// ---------------------------------------------------------------------------
// ComputePrivacyBucketsDelta for MI455X (gfx1250, wave32)
//
// Phases:
//   1) hist:  16M samples -> two 10002-bin f32 histograms (LDS ds_add_f32,
//             merged with global_atomic_add_f32)
//   2) 10x fold: full self-convolution of the 10001-length inner vector via
//             V_WMMA_F32_16X16X4_F32 (tile-pair per wave, shared B operand),
//             plus boundary reductions/assembly
//   3) final: dot with analytic g tensor -> 2 outputs
// ---------------------------------------------------------------------------

typedef __attribute__((ext_vector_type(2))) float v2f;
typedef __attribute__((ext_vector_type(8))) float v8f;

#define BUCKETS_HALF 5000
#define VEC_N        10002   // VECTOR_SIZE
#define INNER_N      10001   // VECTOR_SIZE - 1
#define EXT_N        20001   // EXTENDED
#define N_FOLD       10

// Padded inner buffer: inner[0] lives at PAD_OFF; zeros elsewhere.
#define PAD_OFF      10496
#define PAD_LEN      31232
#define EXT_ALLOC    20480   // 80 tiles * 256; tiles beyond EXT_N land in zeroed pad

// jbb loop: multiples of 4 starting at -240; padded to J_SPLIT*J_CHUNK iters
// (valid range needs 2561; padding reads zero region -> contributes nothing)
#define J_SPLIT      16
#define J_CHUNK      161     // 16*161 = 2576 >= 2561
#define TILE_PAIRS   40      // 80 tiles of 256 outputs, two per wave

// workspace layout (float offsets)
#define WS_PBA   0
#define WS_PBB   10240
#define WS_PADA  20480
#define WS_PADB  (WS_PADA + PAD_LEN)
#define WS_EXTA  (WS_PADB + PAD_LEN)
#define WS_EXTB  (WS_EXTA + EXT_ALLOC)
#define WS_SUMS  (WS_EXTB + EXT_ALLOC)   // 4 floats: {minusA, tailA, minusB, tailB}

__device__ __forceinline__ float ln_f() { return (float)(1.0 / 4000.0); }

// ---------------------------------------------------------------------------
// zero the two distribution vectors
// ---------------------------------------------------------------------------
__global__ void zero_kernel(float* pbA, float* pbB) {
    int i = blockIdx.x * blockDim.x + threadIdx.x;
    if (i < VEC_N) { pbA[i] = 0.0f; pbB[i] = 0.0f; }
}

// ---------------------------------------------------------------------------
// histogram: blockIdx.y selects distribution (0: pb from p_A/pl, 1: dual)
// ---------------------------------------------------------------------------
__global__ __launch_bounds__(256)
void hist_kernel(const float* __restrict__ pA, const float* __restrict__ pB,
                 const float* __restrict__ distA, const float* __restrict__ distB,
                 float* pbA, float* pbB, int n) {
    __shared__ float h[VEC_N];
    for (int i = threadIdx.x; i < VEC_N; i += blockDim.x) h[i] = 0.0f;
    __syncthreads();

    const int d = blockIdx.y;
    const float sgn = d ? -1.0f : 1.0f;
    const int stride = gridDim.x * blockDim.x;
    for (int i = blockIdx.x * blockDim.x + threadIdx.x; i < n; i += stride) {
        float a = pA[i];
        float b = pB[i];
        float pl = logf(a) - logf(b);              // jnp.log in f32
        float q  = (sgn * pl) / ln_f();            // f32 divide, as reference
        int idx  = (int)ceilf(q);
        int pos;
        if (idx > BUCKETS_HALF) pos = VEC_N - 1;
        else {
            idx = idx < -BUCKETS_HALF ? -BUCKETS_HALF : idx;
            idx = idx >  BUCKETS_HALF ?  BUCKETS_HALF : idx;
            pos = idx + BUCKETS_HALF;
        }
        atomicAdd(&h[pos], d ? b : a);             // ds_add_f32
    }
    __syncthreads();

    float* pb = d ? pbB : pbA;
    for (int i = threadIdx.x; i < VEC_N; i += blockDim.x) {
        float v = h[i];
        if (v != 0.0f) atomicAdd(&pb[i], v);       // global_atomic_add_f32
    }
    if (blockIdx.x == 0 && threadIdx.x == 0)
        atomicAdd(&pb[VEC_N - 1], d ? distB[0] : distA[0]);
}

// ---------------------------------------------------------------------------
// prep: build zero-padded inner buffer from pb, zero ext accumulator
// ---------------------------------------------------------------------------
__global__ void prep_kernel(const float* pbA, const float* pbB,
                            float* padA, float* padB,
                            float* extA, float* extB) {
    int i = blockIdx.x * blockDim.x + threadIdx.x;
    const float* pb  = blockIdx.y ? pbB  : pbA;
    float*       pad = blockIdx.y ? padB : padA;
    float*       ext = blockIdx.y ? extB : extA;
    if (i < PAD_LEN) {
        int t = i - PAD_OFF;
        pad[i] = (t >= 0 && t < INNER_N) ? pb[t] : 0.0f;
    }
    if (i < EXT_ALLOC) ext[i] = 0.0f;
}

// ---------------------------------------------------------------------------
// conv: ext = conv(inner, inner) via V_WMMA_F32_16X16X4_F32.
// One wave per (tile-pair, j-chunk); two adjacent 256-output tiles share the
// same B operand.  Output tile: k = k0 + r + 16c.  K-slot s~ carries
//   j(s~) = jbb - s~ + 16c       (free relabeling of K slots)
//   A[r][s~] = pad[PAD_OFF + k0 - jbb + r + s~]   -> ascending pair (no swap)
//   B[s~][c] = pad[PAD_OFF + jbb - s~ + 16c]      -> one reversed pair
// Looping jbb over 4Z covers each j exactly once per column; zero padding
// absorbs out-of-range indices (no double counting, no branches).
// ---------------------------------------------------------------------------
__global__ __launch_bounds__(32)
void conv_kernel(const float* __restrict__ padA, const float* __restrict__ padB,
                 float* __restrict__ extA, float* __restrict__ extB) {
    const int pair = blockIdx.x;                  // 0..TILE_PAIRS-1
    const int jcs  = blockIdx.y;                  // 0..J_SPLIT-1
    const float* pad = blockIdx.z ? padB : padA;
    float*       ext = blockIdx.z ? extB : extA;

    const int k0 = pair * 512;                    // first tile; second at k0+256
    const int L  = threadIdx.x;                   // lane 0..31 (wave32)
    const int r  = L & 15;                        // A row / B,C column
    const int hi = L >> 4;                        // lane half
    const int s0 = hi * 2;                        // K slots s0, s0+1 in this half
    const int bofs = ((L & 15) << 4) - s0;        // B: 16*c - s0

    const int start = jcs * J_CHUNK;
    const int jbb0  = -240 + start * 4;
    int aidx = PAD_OFF + k0 - jbb0 + (r + s0);    // steps -4 per iter
    int bidx = PAD_OFF + jbb0 + bofs;             // steps +4 per iter

    v8f acc0 = {};
    v8f acc1 = {};
    #pragma unroll 2
    for (int t = 0; t < J_CHUNK; ++t) {
        v2f a0, a1, b;
        a0.x = pad[aidx];                         // tile 0: A[r][s0]
        a0.y = pad[aidx + 1];                     // tile 0: A[r][s0+1]  (ascending)
        a1.x = pad[aidx + 256];                   // tile 1: A[r][s0]
        a1.y = pad[aidx + 257];                   // tile 1: A[r][s0+1]  (ascending)
        b.x  = pad[bidx];                         // B[s0][c]
        b.y  = pad[bidx - 1];                     // B[s0+1][c]         (reversed)
        acc0 = __builtin_amdgcn_wmma_f32_16x16x4_f32(
                   false, a0, false, b, (short)0, acc0, false, false);
        acc1 = __builtin_amdgcn_wmma_f32_16x16x4_f32(
                   false, a1, false, b, (short)0, acc1, false, false);
        aidx -= 4;
        bidx += 4;
    }

    // C/D layout: VGPR v, lane L -> M = v + 8*hi, N = L&15 ; k = k0 + M + 16N
    const int kbase = k0 + (hi << 3) + ((L & 15) << 4);
    #pragma unroll
    for (int v = 0; v < 8; ++v)
        atomicAdd(&ext[kbase + v], acc0[v]);          // in-bounds by construction
    #pragma unroll
    for (int v = 0; v < 8; ++v)
        atomicAdd(&ext[kbase + 256 + v], acc1[v]);    // phantom tile hits zeroed pad
}

// ---------------------------------------------------------------------------
// reduce: minus_n = sum(ext[0:5001]);  tail = sum(ext[15001:20001])
// one block per distribution
// ---------------------------------------------------------------------------
__global__ __launch_bounds__(256)
void reduce_kernel(const float* extA, const float* extB, float* sums) {
    const float* ext = blockIdx.x ? extB : extA;
    __shared__ float sm[256];
    const int tid = threadIdx.x;

    float s1 = 0.0f;
    for (int i = tid; i <= BUCKETS_HALF; i += 256) s1 += ext[i];
    sm[tid] = s1; __syncthreads();
    for (int o = 128; o > 0; o >>= 1) {
        if (tid < o) sm[tid] += sm[tid + o];
        __syncthreads();
    }
    if (tid == 0) sums[blockIdx.x * 2] = sm[0];
    __syncthreads();

    float s2 = 0.0f;
    for (int i = EXT_N - BUCKETS_HALF + tid; i < EXT_N; i += 256) s2 += ext[i];
    sm[tid] = s2; __syncthreads();
    for (int o = 128; o > 0; o >>= 1) {
        if (tid < o) sm[tid] += sm[tid + o];
        __syncthreads();
    }
    if (tid == 0) sums[blockIdx.x * 2 + 1] = sm[0];
}

// ---------------------------------------------------------------------------
// assemble: pb <- [minus_n, ext[5001:15001], 2*inf - inf^2 + tail]
// ---------------------------------------------------------------------------
__global__ void assemble_kernel(float* pbA, float* pbB,
                                const float* extA, const float* extB,
                                const float* sums) {
    int i = blockIdx.x * blockDim.x + threadIdx.x;
    const int d = blockIdx.y;
    float*       pb  = d ? pbB  : pbA;
    const float* ext = d ? extB : extA;
    if (i == 0) {
        pb[0] = sums[d * 2];
    } else if (i < INNER_N) {
        pb[i] = ext[BUCKETS_HALF + i];            // ext[5001 + (i-1)]
    } else if (i == INNER_N) {
        float inf_b = pb[INNER_N];
        inf_b = inf_b + inf_b - inf_b * inf_b;
        pb[INNER_N] = inf_b + sums[d * 2 + 1];
    }
}

// ---------------------------------------------------------------------------
// final: out[d] = sum_{i=9001}^{10000} pb[i]*(1 - exp(-LN_F*(i-9000))) + pb[10001]
// ---------------------------------------------------------------------------
__global__ __launch_bounds__(256)
void final_kernel(const float* pbA, const float* pbB, float* out) {
    const int d = blockIdx.x;
    const float* pb = d ? pbB : pbA;
    __shared__ float sm[256];
    const int tid = threadIdx.x;
    const int gstart = BUCKETS_HALF + 4000 + 1;          // 9001
    float s = 0.0f;
    for (int i = gstart + tid; i <= 2 * BUCKETS_HALF; i += 256) {
        int j = i - (gstart - 1);                        // 1..1000
        float g = (float)(1.0 - exp(-(double)j * (1.0 / 4000.0)));
        s += pb[i] * g;
    }
    sm[tid] = s; __syncthreads();
    for (int o = 128; o > 0; o >>= 1) {
        if (tid < o) sm[tid] += sm[tid + o];
        __syncthreads();
    }
    if (tid == 0) out[d] = sm[0] + pb[VEC_N - 1];        // g[-1] = 1
}

// ---------------------------------------------------------------------------
extern "C" void kernel_launch(void* const* d_in, const int* in_sizes, int n_in,
                              void* d_out, int out_size, void* d_ws, size_t ws_size,
                              hipStream_t stream) {
    const float* pA    = (const float*)d_in[0];
    const float* pB    = (const float*)d_in[1];
    const float* distA = (const float*)d_in[2];
    const float* distB = (const float*)d_in[3];
    const int n = in_sizes[0];

    float* ws   = (float*)d_ws;
    float* pbA  = ws + WS_PBA;
    float* pbB  = ws + WS_PBB;
    float* padA = ws + WS_PADA;
    float* padB = ws + WS_PADB;
    float* extA = ws + WS_EXTA;
    float* extB = ws + WS_EXTB;
    float* sums = ws + WS_SUMS;

    // 1) zero distributions, then histogram both
    zero_kernel<<<dim3((VEC_N + 255) / 256), 256, 0, stream>>>(pbA, pbB);
    hist_kernel<<<dim3(2048, 2), 256, 0, stream>>>(pA, pB, distA, distB, pbA, pbB, n);

    // 2) ten self-convolution folds
    for (int it = 0; it < N_FOLD; ++it) {
        prep_kernel<<<dim3((PAD_LEN + 255) / 256, 2), 256, 0, stream>>>(
            pbA, pbB, padA, padB, extA, extB);
        conv_kernel<<<dim3(TILE_PAIRS, J_SPLIT, 2), 32, 0, stream>>>(
            padA, padB, extA, extB);
        reduce_kernel<<<2, 256, 0, stream>>>(extA, extB, sums);
        assemble_kernel<<<dim3((VEC_N + 255) / 256, 2), 256, 0, stream>>>(
            pbA, pbB, extA, extB, sums);
    }

    // 3) dot with g tensor
    final_kernel<<<2, 256, 0, stream>>>(pbA, pbB, (float*)d_out);
}